// SymbolicLogic_17978733101822
// MI455X (gfx1250) — compile-verified
//
#include <hip/hip_runtime.h>

// Problem constants (reference: N=131072, L=81, C=10, float32)
constexpr int kL = 81;
constexpr int kC = 10;
constexpr int kRowFloats = kL * kC;                       // 810
constexpr int kRowsPerBlock = 8;
constexpr int kThreads = 256;                             // 8 waves (wave32)
constexpr int kBlockFloats = kRowsPerBlock * kRowFloats;  // 6480
constexpr int kBlockBytes = kBlockFloats * 4;             // 25920 (16B divisible)
constexpr int kChunks = kBlockBytes / 16;                 // 1620 x b128
constexpr int kRounds = (kChunks + kThreads - 1) / kThreads;  // 7

// The async-LDS builtins take pointers to 16B int vectors in AS(1)/AS(3).
typedef __attribute__((ext_vector_type(4))) int v4i;
typedef __attribute__((address_space(1))) v4i gas_v4i;
typedef __attribute__((address_space(3))) v4i las_v4i;

__device__ __forceinline__ void async_g2l_b128(const void* gsrc, void* ldst) {
#if __has_builtin(__builtin_amdgcn_global_load_async_to_lds_b128)
  __builtin_amdgcn_global_load_async_to_lds_b128(
      (gas_v4i*)gsrc, (las_v4i*)ldst, /*offset=*/0, /*cpol=*/0);
#else
  unsigned lds_addr = (unsigned)(unsigned long long)(las_v4i*)ldst;
  unsigned long long ga = (unsigned long long)gsrc;
  asm volatile("global_load_async_to_lds_b128 %0, %1, off"
               :: "v"(lds_addr), "v"(ga) : "memory");
#endif
}

__device__ __forceinline__ void async_l2g_b128(void* gdst, const void* lsrc) {
#if __has_builtin(__builtin_amdgcn_global_store_async_from_lds_b128)
  __builtin_amdgcn_global_store_async_from_lds_b128(
      (gas_v4i*)gdst, (las_v4i*)lsrc, /*offset=*/0, /*cpol=*/0);
#else
  unsigned lds_addr = (unsigned)(unsigned long long)(las_v4i*)lsrc;
  unsigned long long ga = (unsigned long long)gdst;
  asm volatile("global_store_async_from_lds_b128 %1, %0, off"
               :: "v"(lds_addr), "v"(ga) : "memory");
#endif
}

__device__ __forceinline__ void wait_async0() {
#if __has_builtin(__builtin_amdgcn_s_wait_asynccnt)
  __builtin_amdgcn_s_wait_asynccnt(0);
#else
  asm volatile("s_wait_asynccnt 0" ::: "memory");
#endif
}

__global__ __launch_bounds__(kThreads) void sudoku_fill_kernel(
    const float* __restrict__ in, float* __restrict__ out) {
  __shared__ __align__(16) float buf[kBlockFloats];

  const int tid = threadIdx.x;
  const long long base = (long long)blockIdx.x * kBlockFloats;  // in floats
  const char* gin = (const char*)(in + base);
  char* gout = (char*)(out + base);
  char* lbase = (char*)buf;

  // ---- Phase 1: async DMA global -> LDS (8 rows = 25,920 B) ----
#pragma unroll
  for (int i = 0; i < kRounds; ++i) {
    int idx = i * kThreads + tid;
    if (idx < kChunks) {
      async_g2l_b128(gin + (long long)idx * 16, lbase + idx * 16);
    }
  }
  wait_async0();
  __syncthreads();

  // ---- Phase 2: per-row argmax + presence mask (one wave32 per row) ----
  const int wave = tid >> 5;
  const int lane = tid & 31;
  float* row = buf + wave * kRowFloats;

  unsigned mask = 0u;
  for (int l = lane; l < kL; l += 32) {
    const float2* p = (const float2*)(row + l * kC);  // 8B aligned always
    float2 v0 = p[0], v1 = p[1], v2 = p[2], v3 = p[3], v4 = p[4];
    float vals[kC] = {v0.x, v0.y, v1.x, v1.y, v2.x,
                      v2.y, v3.x, v3.y, v4.x, v4.y};
    int best = 0;
    float bv = vals[0];
#pragma unroll
    for (int c = 1; c < kC; ++c) {
      if (vals[c] > bv) {  // strict > keeps first-index argmax semantics
        bv = vals[c];
        best = c;
      }
    }
    mask |= 1u << best;
  }
  // wave32 OR-reduction: every lane ends with the row's full presence mask
#pragma unroll
  for (int s = 1; s < 32; s <<= 1) {
    mask |= __shfl_xor(mask, s, 32);
  }

  // ---- Phase 3: patch positions 1..9 in LDS if digit (l-1) is missing ----
  if (lane >= 1 && lane <= 9) {
    const int j = lane - 1;
    if (((mask >> j) & 1u) == 0u) {
      float2* p = (float2*)(row + lane * kC);
#pragma unroll
      for (int k = 0; k < 5; ++k) {
        float2 w;
        w.x = (2 * k == j) ? 1.0f : 0.0f;
        w.y = (2 * k + 1 == j) ? 1.0f : 0.0f;
        p[k] = w;
      }
    }
  }
  __syncthreads();

  // ---- Phase 4: async DMA LDS -> global output (bulk copy incl. patches) ----
#pragma unroll
  for (int i = 0; i < kRounds; ++i) {
    int idx = i * kThreads + tid;
    if (idx < kChunks) {
      async_l2g_b128(gout + (long long)idx * 16, lbase + idx * 16);
    }
  }
  // S_ENDPGM performs an implicit wait-idle, draining outstanding async stores.
}

extern "C" void kernel_launch(void* const* d_in, const int* in_sizes, int n_in,
                              void* d_out, int out_size, void* d_ws, size_t ws_size,
                              hipStream_t stream) {
  (void)n_in; (void)out_size; (void)d_ws; (void)ws_size;
  const float* in = (const float*)d_in[0];
  float* out = (float*)d_out;
  const int n_rows = in_sizes[0] / kRowFloats;       // 131072
  const int blocks = n_rows / kRowsPerBlock;         // 16384
  hipLaunchKernelGGL(sudoku_fill_kernel, dim3(blocks), dim3(kThreads), 0, stream,
                     in, out);
}